// Detect_74036646248814
// MI455X (gfx1250) — compile-verified
//
#include <hip/hip_runtime.h>
#include <cstdint>
#include <cstddef>

typedef __attribute__((ext_vector_type(2))) float v2f;
typedef __attribute__((ext_vector_type(8))) float v8f;

constexpr int   BATCH = 8;
constexpr int   ANCH  = 16320;
constexpr int   NCLS  = 81;     // incl. background
constexpr int   TOPK  = 200;
constexpr float CONF_T = 0.01f;
constexpr float NMS_T  = 0.45f;
constexpr float VAR0   = 0.1f;
constexpr float VAR1   = 0.2f;

// ---------------------------------------------------------------- zero output
__global__ void k_zero(float* __restrict__ o, int n) {
    int i = blockIdx.x * blockDim.x + threadIdx.x;
    if (i < n) o[i] = 0.f;
}

// ------------------------------------------------- stage 1: softmax+mask -> T
// scoresT layout: [b][c0..79][a], c stores foreground class (c+1) scores.
__global__ __launch_bounds__(256)
void k_scores(const float* __restrict__ conf, const int* __restrict__ ign,
              float* __restrict__ sT) {
    int g = blockIdx.x * 256 + threadIdx.x;
    if (g >= BATCH * ANCH) return;
    const float* row = conf + (size_t)g * NCLS;
    __builtin_prefetch(row, 0, 0);            // global_prefetch_b8
    float m = row[0];
#pragma unroll 1
    for (int c = 1; c < NCLS; ++c) m = fmaxf(m, row[c]);
    float d = 0.f;
#pragma unroll 1
    for (int c = 0; c < NCLS; ++c) d += expf(row[c] - m);
    float inv = 1.f / d;
    bool valid = ign[g] < 1;
    int b = g / ANCH, a = g - b * ANCH;
    float* op = sT + (size_t)b * 80 * ANCH + a;
#pragma unroll 1
    for (int c = 1; c < NCLS; ++c) {
        float s = expf(row[c] - m) * inv;
        s = (valid && s > CONF_T) ? s : 0.f;
        op[(size_t)(c - 1) * ANCH] = s;       // coalesced across threads
    }
}

// -------------------------------------------- stage 2: top-K + IoU(WMMA) + NMS
__global__ __launch_bounds__(256)
void k_detect(const float* __restrict__ loc, const float* __restrict__ conf,
              const float* __restrict__ anchors, const int* __restrict__ ign,
              const float* __restrict__ sT, int useT, float* __restrict__ out) {
    __shared__ float    s_vals[TOPK];
    __shared__ int      s_sel[TOPK];
    __shared__ float    s_cand[208 * 4];      // padded to 13 WMMA tiles
    __shared__ float    s_area[208];
    __shared__ unsigned s_sup[TOPK * 7];      // 224-bit suppression mask per row
    __shared__ unsigned s_keep[7];
    __shared__ float    s_ws[8];
    __shared__ int      s_wi[8];
    __shared__ int      s_win_i;

    const int tid  = threadIdx.x;
    const int lane = tid & 31;
    const int wid  = tid >> 5;
    const int blk  = blockIdx.x;
    const int b    = blk / 80;
    const int c    = blk % 80;                // output class = c+1

    // ---- load masked scores into registers: thread t owns a = j*256 + t ----
    float sc[64];
    const float* col = sT + ((size_t)(b * 80 + c)) * ANCH;
    if (useT) __builtin_prefetch(col + tid, 0, 0);
#pragma unroll
    for (int j = 0; j < 64; ++j) {
        int a = j * 256 + tid;
        float v = -1.f;                       // consumed/padding sentinel
        if (a < ANCH) {
            if (useT) {
                v = col[a];
            } else {
                const float* row = conf + ((size_t)b * ANCH + a) * NCLS;
                float m = row[0];
#pragma unroll 1
                for (int cc = 1; cc < NCLS; ++cc) m = fmaxf(m, row[cc]);
                float d = 0.f;
#pragma unroll 1
                for (int cc = 0; cc < NCLS; ++cc) d += expf(row[cc] - m);
                float s = expf(row[c + 1] - m) / d;
                v = (ign[(size_t)b * ANCH + a] < 1 && s > CONF_T) ? s : 0.f;
            }
        }
        sc[j] = v;
    }

    // ---- initial per-thread best ----
    uint64_t cons = 0;
    float ls = -2.f; int li = 0x3FFFFFFF;
#pragma unroll
    for (int j = 0; j < 64; ++j) {
        int a = j * 256 + tid;
        if (sc[j] > ls || (sc[j] == ls && a < li)) { ls = sc[j]; li = a; }
    }

    // ---- 200 rounds of incremental block-argmax (tie -> lowest index) ----
    for (int k = 0; k < TOPK; ++k) {
        float s = ls; int i = li;
#pragma unroll
        for (int off = 16; off > 0; off >>= 1) {        // wave32 reduce
            float os = __shfl_xor(s, off);
            int   oi = __shfl_xor(i, off);
            if (os > s || (os == s && oi < i)) { s = os; i = oi; }
        }
        if (lane == 0) { s_ws[wid] = s; s_wi[wid] = i; }
        __syncthreads();
        if (wid == 0) {
            float s2 = (lane < 8) ? s_ws[lane] : -3.f;
            int   i2 = (lane < 8) ? s_wi[lane] : 0x3FFFFFFF;
#pragma unroll
            for (int off = 4; off > 0; off >>= 1) {
                float os = __shfl_xor(s2, off);
                int   oi = __shfl_xor(i2, off);
                if (os > s2 || (os == s2 && oi < i2)) { s2 = os; i2 = oi; }
            }
            if (lane == 0) { s_win_i = i2; s_vals[k] = s2; s_sel[k] = i2; }
        }
        __syncthreads();
        int wi = s_win_i;
        if ((wi & 255) == tid) {              // only the owner rescans its 64
            cons |= 1ull << (wi >> 8);
            ls = -2.f; li = 0x3FFFFFFF;
#pragma unroll
            for (int j = 0; j < 64; ++j) {
                if (!((cons >> j) & 1ull)) {
                    int a = j * 256 + tid;
                    if (sc[j] > ls || (sc[j] == ls && a < li)) { ls = sc[j]; li = a; }
                }
            }
        }
    }
    __syncthreads();

    // ---- clear NMS scratch ----
    for (int i = tid; i < 208 * 4; i += 256) s_cand[i] = 0.f;
    for (int i = tid; i < 208;     i += 256) s_area[i] = 0.f;
    for (int i = tid; i < TOPK * 7; i += 256) s_sup[i] = 0u;
    if (tid < 7) s_keep[tid] = 0u;
    __syncthreads();

    // ---- decode the 200 candidate boxes ----
    if (tid < TOPK) {
        int g = s_sel[tid];
        const float* lp = loc     + ((size_t)b * ANCH + g) * 4;
        const float* pp = anchors + ((size_t)b * ANCH + g) * 4;
        float cx = pp[0] + lp[0] * VAR0 * pp[2];
        float cy = pp[1] + lp[1] * VAR0 * pp[3];
        float w  = pp[2] * expf(lp[2] * VAR1);
        float h  = pp[3] * expf(lp[3] * VAR1);
        float x1 = cx - w * 0.5f, y1 = cy - h * 0.5f;
        float x2 = cx + w * 0.5f, y2 = cy + h * 0.5f;
        s_cand[tid * 4 + 0] = x1; s_cand[tid * 4 + 1] = y1;
        s_cand[tid * 4 + 2] = x2; s_cand[tid * 4 + 3] = y2;
        s_area[tid] = (x2 - x1) * (y2 - y1);
        if (s_vals[tid] > 0.f) atomicOr(&s_keep[tid >> 5], 1u << (tid & 31));
    }
    __syncthreads();

    // ---- 16x16 IoU tiles: area_i + area_j via WMMA outer products ----
    // C/D layout: lane l, vgpr r -> (M = r + 8*(l>>4), N = l&15)
    for (int t = wid; t < 13 * 13; t += 8) {          // uniform per wave
        int i0 = (t / 13) * 16, j0 = (t % 13) * 16;
        bool hi = lane >= 16; int lo = lane & 15;
        v8f acc;
#if __has_builtin(__builtin_amdgcn_wmma_f32_16x16x4_f32)
        // A(16x4): lanes0-15 hold K=0,1; lanes16-31 hold K=2,3 (set to 0)
        v2f A1, B1, A2, B2;
        A1.x = hi ? 0.f : s_area[i0 + lo]; A1.y = 0.f;   // col of areas (K=0)
        B1.x = hi ? 0.f : 1.f;             B1.y = 0.f;   // row of ones  (K=0)
        A2.x = hi ? 0.f : 1.f;             A2.y = 0.f;
        B2.x = hi ? 0.f : s_area[j0 + lo]; B2.y = 0.f;
        v8f cz = (v8f){0.f, 0.f, 0.f, 0.f, 0.f, 0.f, 0.f, 0.f};
        acc = __builtin_amdgcn_wmma_f32_16x16x4_f32(false, A1, false, B1,
                                                    (short)0, cz, false, false);
        acc = __builtin_amdgcn_wmma_f32_16x16x4_f32(false, A2, false, B2,
                                                    (short)0, acc, false, false);
#else
#pragma unroll
        for (int r = 0; r < 8; ++r)
            acc[r] = s_area[i0 + r + (hi ? 8 : 0)] + s_area[j0 + lo];
#endif
        int n = j0 + lo;
        float xn1 = s_cand[n * 4 + 0], yn1 = s_cand[n * 4 + 1];
        float xn2 = s_cand[n * 4 + 2], yn2 = s_cand[n * 4 + 3];
#pragma unroll
        for (int r = 0; r < 8; ++r) {
            int m = i0 + r + (hi ? 8 : 0);
            float xm1 = s_cand[m * 4 + 0], ym1 = s_cand[m * 4 + 1];
            float xm2 = s_cand[m * 4 + 2], ym2 = s_cand[m * 4 + 3];
            float iw = fmaxf(fminf(xm2, xn2) - fmaxf(xm1, xn1), 0.f);
            float ih = fmaxf(fminf(ym2, yn2) - fmaxf(ym1, yn1), 0.f);
            float inter = iw * ih;
            float uni = acc[r] - inter;       // area_i + area_j - inter
            if (inter > NMS_T * uni && m < TOPK && n < TOPK)
                atomicOr(&s_sup[m * 7 + (n >> 5)], 1u << (n & 31));
        }
    }
    __syncthreads();

    // ---- sequential suppression (matches fori_loop exactly) ----
    if (tid == 0) {
        unsigned kp[7];
#pragma unroll
        for (int w = 0; w < 7; ++w) kp[w] = s_keep[w];
        for (int i = 0; i < TOPK; ++i) {
            int w = i >> 5, bt = i & 31;
            if ((kp[w] >> bt) & 1u) {
                unsigned gtm = (bt == 31) ? 0u : (0xFFFFFFFFu << (bt + 1));
                kp[w] &= ~(s_sup[i * 7 + w] & gtm);     // only j > i
                for (int ww = w + 1; ww < 7; ++ww) kp[ww] &= ~s_sup[i * 7 + ww];
            }
        }
#pragma unroll
        for (int w = 0; w < 7; ++w) s_keep[w] = kp[w];
    }
    __syncthreads();

    // ---- compaction + write (out is pre-zeroed) ----
    if (tid < TOPK) {
        int w = tid >> 5, bt = tid & 31;
        if ((s_keep[w] >> bt) & 1u) {
            int p = 0;
            for (int ww = 0; ww < w; ++ww) p += __popc(s_keep[ww]);
            p += __popc(s_keep[w] & ((1u << bt) - 1u));
            float* o = out + ((((size_t)b * NCLS) + (c + 1)) * TOPK + p) * 5;
            o[0] = s_vals[tid];
            o[1] = s_cand[tid * 4 + 0];
            o[2] = s_cand[tid * 4 + 1];
            o[3] = s_cand[tid * 4 + 2];
            o[4] = s_cand[tid * 4 + 3];
        }
    }
}

// --------------------------------------------------------------------- launch
extern "C" void kernel_launch(void* const* d_in, const int* in_sizes, int n_in,
                              void* d_out, int out_size, void* d_ws,
                              size_t ws_size, hipStream_t stream) {
    (void)in_sizes; (void)n_in;
    const float* loc     = (const float*)d_in[0];
    const float* conf    = (const float*)d_in[1];
    const float* anchors = (const float*)d_in[2];
    const int*   ign     = (const int*)d_in[3];
    float*       out     = (float*)d_out;
    float*       sT      = (float*)d_ws;

    const size_t needT = (size_t)BATCH * 80 * ANCH * sizeof(float);
    int useT = (d_ws != nullptr && ws_size >= needT) ? 1 : 0;

    int nOut = out_size;                       // 8*81*200*5 = 648000
    k_zero<<<(nOut + 255) / 256, 256, 0, stream>>>(out, nOut);

    if (useT)
        k_scores<<<(BATCH * ANCH) / 256, 256, 0, stream>>>(conf, ign, sT);

    k_detect<<<BATCH * 80, 256, 0, stream>>>(loc, conf, anchors, ign,
                                             sT, useT, out);
}